// MultiAggregation_67095979098787
// MI455X (gfx1250) — compile-verified
//
#include <hip/hip_runtime.h>
#include <hip/hip_bf16.h>
#include <math.h>

// MultiAggregation (sum / mean / max / std) scatter: 1.6M edges x 64 f32 -> 100K nodes.
// Single pass: sum, sum-of-squares, max, count via native L2 atomics
// (global_atomic_add_f32 / global_atomic_max_num_f32). Message stream staged
// through LDS with gfx1250 async global->LDS loads, double-buffered (ASYNCcnt).

#define D_FEAT 64
#define EDGES_PER_BLOCK 64
#define CHUNK_EDGES 16
#define NCHUNKS (EDGES_PER_BLOCK / CHUNK_EDGES)

typedef float v4f __attribute__((ext_vector_type(4)));
typedef int v4i __attribute__((ext_vector_type(4)));
typedef __attribute__((address_space(1))) v4i* as1_v4i_ptr;  // global
typedef __attribute__((address_space(3))) v4i* as3_v4i_ptr;  // LDS

#ifndef __has_builtin
#define __has_builtin(x) 0
#endif

#if __has_builtin(__builtin_amdgcn_global_load_async_to_lds_b128)
#define HAVE_ASYNC_LDS 1
#else
#define HAVE_ASYNC_LDS 0
#endif

// Verified (round 3 vs 4): with this attribute the atomicrmw fadd lowers to a
// single native global_atomic_add_f32 (vmem count identical to inline asm),
// while keeping compiler address/offset folding.
#if defined(__has_cpp_attribute)
#if __has_cpp_attribute(clang::atomic)
#define FAST_FP_ATOMICS \
  [[clang::atomic(no_remote_memory, no_fine_grained_memory, ignore_denormal_mode)]]
#endif
#endif
#ifndef FAST_FP_ATOMICS
#define FAST_FP_ATOMICS
#endif

__device__ __forceinline__ void atomic_add_f32(float* p, float v) {
  FAST_FP_ATOMICS {
    (void)__hip_atomic_fetch_add(p, v, __ATOMIC_RELAXED, __HIP_MEMORY_SCOPE_AGENT);
  }
}

__device__ __forceinline__ void atomic_max_f32(float* p, float v) {
#if defined(__gfx1250__)
  // Native IEEE maximumNumber atomic: one instruction, branchless, handles
  // both signs and the -inf init directly.
  asm volatile("global_atomic_max_num_f32 %0, %1, off scope:SCOPE_DEV"
               :
               : "v"(p), "v"(v)
               : "memory");
#else
  if (v >= 0.0f) {
    atomicMax((int*)p, __float_as_int(v));
  } else {
    atomicMin((unsigned int*)p, __float_as_uint(v));
  }
#endif
}

template <int N>
__device__ __forceinline__ void wait_asynccnt() {
#if __has_builtin(__builtin_amdgcn_s_wait_asynccnt)
  __builtin_amdgcn_s_wait_asynccnt(N);
#else
  if constexpr (N == 0)
    asm volatile("s_wait_asynccnt 0x0" ::: "memory");
  else
    asm volatile("s_wait_asynccnt 0x1" ::: "memory");
#endif
  asm volatile("" ::: "memory");  // keep LDS reads after the wait
}

#if HAVE_ASYNC_LDS
__device__ __forceinline__ void async_load_16B(const float* gp, v4f* lp) {
  __builtin_amdgcn_global_load_async_to_lds_b128(
      (as1_v4i_ptr)gp, (as3_v4i_ptr)lp, 0, 0);
}
#endif

// --- Kernel 1: initialize accumulators, 4-wide (b128 stores). Required every
// call: the harness does not re-poison buffers between graph replays.
__global__ void ma_init_kernel(float* __restrict__ out, float* __restrict__ sumsq,
                               float* __restrict__ cnt, int N) {
  int idx = blockIdx.x * blockDim.x + threadIdx.x;  // over N*16
  if (idx >= N * (D_FEAT / 4)) return;
  int n = idx >> 4;
  int c4 = (idx & 15) << 2;
  const v4f z = {0.0f, 0.0f, 0.0f, 0.0f};
  const v4f ninf = {-INFINITY, -INFINITY, -INFINITY, -INFINITY};
  *(v4f*)(out + (size_t)n * 256 + c4) = z;           // sum region
  *(v4f*)(out + (size_t)n * 256 + 128 + c4) = ninf;  // max region
  *(v4f*)(sumsq + (size_t)n * D_FEAT + c4) = z;
  if ((idx & 15) == 0) cnt[n] = 0.0f;
}

// --- Kernel 2: edge-parallel scatter. Block = 64 edges, 4 pipelined stages of
// 16 edges. Thread t stages/consumes its own 16B (edge = t>>4, feats (t&15)*4).
__global__ void ma_scatter_kernel(const float* __restrict__ messages,
                                  const int* __restrict__ dst,
                                  float* __restrict__ out,
                                  float* __restrict__ sumsq,
                                  float* __restrict__ cnt,
                                  int E) {
  const int tid = threadIdx.x;
  const long base = (long)blockIdx.x * EDGES_PER_BLOCK;
  const int c4 = (tid & 15) << 2;

#if HAVE_ASYNC_LDS
  __shared__ v4f buf[2][256];  // 2 x 4KB double buffer
  const long maxElem = (long)E * D_FEAT - 4;  // clamp keeps async loads in-bounds

  {  // prologue: issue chunk 0
    long elem = base * D_FEAT + (long)tid * 4;
    if (elem > maxElem) elem = maxElem;
    async_load_16B(messages + elem, &buf[0][tid]);
  }
#pragma unroll
  for (int c = 0; c < NCHUNKS; ++c) {
    if (c + 1 < NCHUNKS) {  // issue next chunk, then wait for current (<=1 in flight)
      long elem = (base + (long)(c + 1) * CHUNK_EDGES) * D_FEAT + (long)tid * 4;
      if (elem > maxElem) elem = maxElem;
      async_load_16B(messages + elem, &buf[(c + 1) & 1][tid]);
      wait_asynccnt<1>();
    } else {
      wait_asynccnt<0>();
    }
    v4f m = buf[c & 1][tid];  // ds_load_b128; own bytes only -> no barrier

    long e = base + (long)c * CHUNK_EDGES + (tid >> 4);
    if (e < E) {
      int d = dst[e];
      float* ps = out + (size_t)d * 256 + c4;       // sum
      float* pm = ps + 128;                         // max
      float* pq = sumsq + (size_t)d * D_FEAT + c4;  // sum of squares
#pragma unroll
      for (int j = 0; j < 4; ++j) {
        float x = m[j];
        atomic_add_f32(ps + j, x);
        atomic_add_f32(pq + j, x * x);
        atomic_max_f32(pm + j, x);
      }
      if ((tid & 15) == 0) atomic_add_f32(cnt + d, 1.0f);
    }
  }
#else
  // Fallback: direct non-temporal loads, same geometry.
  __builtin_prefetch(messages + (base + EDGES_PER_BLOCK) * D_FEAT + tid * 4, 0, 0);
#pragma unroll
  for (int c = 0; c < NCHUNKS; ++c) {
    long e = base + (long)c * CHUNK_EDGES + (tid >> 4);
    if (e < E) {
      v4f m = __builtin_nontemporal_load(
          (const v4f*)(messages + e * D_FEAT + c4));
      int d = dst[e];
      float* ps = out + (size_t)d * 256 + c4;
      float* pm = ps + 128;
      float* pq = sumsq + (size_t)d * D_FEAT + c4;
#pragma unroll
      for (int j = 0; j < 4; ++j) {
        float x = m[j];
        atomic_add_f32(ps + j, x);
        atomic_add_f32(pq + j, x * x);
        atomic_max_f32(pm + j, x);
      }
      if ((tid & 15) == 0) atomic_add_f32(cnt + d, 1.0f);
    }
  }
#endif
}

// --- Kernel 3: finalize mean / std, fix empty-segment max, 4-wide b128 I/O.
__global__ void ma_finalize_kernel(float* __restrict__ out,
                                   const float* __restrict__ sumsq,
                                   const float* __restrict__ cnt, int N) {
  int idx = blockIdx.x * blockDim.x + threadIdx.x;  // over N*16
  if (idx >= N * (D_FEAT / 4)) return;
  int n = idx >> 4;
  int c4 = (idx & 15) << 2;
  float* row = out + (size_t)n * 256;
  v4f s = *(const v4f*)(row + c4);
  v4f mx = *(const v4f*)(row + 128 + c4);
  v4f q = *(const v4f*)(sumsq + (size_t)n * D_FEAT + c4);
  float deg = fmaxf(cnt[n], 1.0f);
  float inv = 1.0f / deg;
  v4f mean, mfix, sd;
#pragma unroll
  for (int j = 0; j < 4; ++j) {
    float mj = s[j] * inv;
    float var = fmaxf(q[j] * inv - mj * mj, 0.0f);
    mean[j] = mj;
    sd[j] = sqrtf(var + 1e-8f);
    mfix[j] = (mx[j] == -INFINITY) ? 0.0f : mx[j];
  }
  *(v4f*)(row + 64 + c4) = mean;   // mean
  *(v4f*)(row + 128 + c4) = mfix;  // max (empty segments -> 0)
  *(v4f*)(row + 192 + c4) = sd;    // std
}

extern "C" void kernel_launch(void* const* d_in, const int* in_sizes, int n_in,
                              void* d_out, int out_size, void* d_ws, size_t ws_size,
                              hipStream_t stream) {
  const float* messages = (const float*)d_in[0];
  const int* dst = (const int*)d_in[1];
  int E = in_sizes[1];                 // number of edges
  int N = out_size / (4 * D_FEAT);     // number of nodes

  float* out = (float*)d_out;
  float* sumsq = (float*)d_ws;                 // N*64 floats (25.6 MB)
  float* cnt = sumsq + (size_t)N * D_FEAT;     // N floats

  int nodeThreads4 = N * (D_FEAT / 4);
  int nodeBlocks = (nodeThreads4 + 255) / 256;
  ma_init_kernel<<<nodeBlocks, 256, 0, stream>>>(out, sumsq, cnt, N);

  int edgeBlocks = (E + EDGES_PER_BLOCK - 1) / EDGES_PER_BLOCK;
  ma_scatter_kernel<<<edgeBlocks, 256, 0, stream>>>(messages, dst, out, sumsq, cnt, E);

  ma_finalize_kernel<<<nodeBlocks, 256, 0, stream>>>(out, sumsq, cnt, N);
}